// KDEDensityBranch_73967926771832
// MI455X (gfx1250) — compile-verified
//
#include <hip/hip_runtime.h>

typedef __attribute__((ext_vector_type(2))) float v2f;
typedef __attribute__((ext_vector_type(4))) float f4;
typedef __attribute__((ext_vector_type(8))) float v8f;

namespace {
constexpr int NXg = 432, NYg = 496;          // density grid
constexpr int Bn = 4, Hc = 248, Wc = 216;    // batch, backbone H/W
constexpr int HW = Hc * Wc;                  // 53568
constexpr int CIN = 384, NDF = 16, NPTS = 400000;
constexpr float XMIN = 0.0f, YMIN = -39.68f, VXs = 0.16f, VYs = 0.16f;
constexpr float SIGMA = 6.25f;               // bandwidth / voxel
constexpr float BNEPS = 1e-3f;

// workspace layout (float offsets)
constexpr size_t GW_O   = 0;     // 15 gaussian weights
constexpr size_t MAX_O  = 16;    // 4 per-batch max (uint bits)
constexpr size_t ST1S_O = 32;    // conv1 sums (8)
constexpr size_t ST1Q_O = 40;    // conv1 sumsq (8)
constexpr size_t ST2S_O = 64;    // conv2 sums (16)
constexpr size_t ST2Q_O = 80;    // conv2 sumsq (16)
constexpr size_t HIST_O = 128;
constexpr size_t GRID3  = (size_t)Bn * NYg * NXg;          // 857088
constexpr size_t TMP_O  = HIST_O + GRID3;
constexpr size_t DM_O   = TMP_O + GRID3;
constexpr size_t DMR_O  = DM_O + GRID3;                    // resized dm (B*HW)
constexpr size_t OUT1_O = HIST_O;                          // reuse hist+tmp (exactly 8*B*HW)
constexpr size_t H1_O   = DMR_O + (size_t)Bn * HW;
constexpr size_t OUT2_O = H1_O + (size_t)Bn * 8 * HW;

// conv2 LDS tile geometry: 8 ch x 3 rows, cols cover global x in [-1, 224] (226 cols)
constexpr int LDSW = 232;       // padded row stride (floats)
constexpr int LCOLS = 226;      // valid staged cols
constexpr int TILES_X = 14;     // ceil(216/16)
}

// ---------------- init: zero header + histogram region ----------------
__global__ void k_zero(float* ws) {
  size_t n = HIST_O + GRID3;
  for (size_t i = blockIdx.x * (size_t)blockDim.x + threadIdx.x; i < n;
       i += (size_t)gridDim.x * blockDim.x)
    ws[i] = 0.0f;
}

__global__ void k_gauss_weights(float* gw) {
  float g[15], s = 0.f;
  for (int t = 0; t < 15; ++t) {
    float c = (float)(t - 7);
    g[t] = expf(-c * c / (2.0f * SIGMA * SIGMA));
    s += g[t];
  }
  for (int t = 0; t < 15; ++t) gw[t] = g[t] / s;
}

// ---------------- histogram of points ----------------
__global__ void k_hist(const float* __restrict__ pts, float* __restrict__ hist) {
  int i = blockIdx.x * blockDim.x + threadIdx.x;
  if (i >= NPTS) return;
  const float* p = pts + (size_t)i * 5;
  int b  = (int)p[0];
  int xi = (int)((p[1] - XMIN) / VXs);
  int yi = (int)((p[2] - YMIN) / VYs);
  xi = min(max(xi, 0), NXg - 1);
  yi = min(max(yi, 0), NYg - 1);
  atomicAdd(&hist[((size_t)b * NYg + yi) * NXg + xi], 1.0f);
}

// ---------------- separable gaussian blur ----------------
__global__ void k_blur_y(const float* __restrict__ src, float* __restrict__ dst,
                         const float* __restrict__ gw) {
  size_t idx = blockIdx.x * (size_t)blockDim.x + threadIdx.x;
  if (idx >= GRID3) return;
  int x = (int)(idx % NXg);
  int y = (int)((idx / NXg) % NYg);
  size_t bbase = (idx / ((size_t)NXg * NYg)) * (size_t)NXg * NYg;
  float s = 0.f;
#pragma unroll
  for (int t = 0; t < 15; ++t) {
    int yy = y + t - 7;
    if (yy >= 0 && yy < NYg) s += gw[t] * src[bbase + (size_t)yy * NXg + x];
  }
  dst[idx] = s;
}

__global__ void k_blur_x(const float* __restrict__ src, float* __restrict__ dst,
                         const float* __restrict__ gw) {
  size_t idx = blockIdx.x * (size_t)blockDim.x + threadIdx.x;
  if (idx >= GRID3) return;
  int x = (int)(idx % NXg);
  size_t rbase = idx - x;
  float s = 0.f;
#pragma unroll
  for (int t = 0; t < 15; ++t) {
    int xx = x + t - 7;
    if (xx >= 0 && xx < NXg) s += gw[t] * src[rbase + xx];
  }
  dst[idx] = s;
}

// ---------------- per-batch max (values >= 0 -> uint compare ok) ----------------
__global__ void k_max(const float* __restrict__ dm, unsigned* __restrict__ mx) {
  int b = blockIdx.x >> 6, s = blockIdx.x & 63;
  const float* base = dm + (size_t)b * NYg * NXg;
  float m = 0.f;
  for (int i = s * blockDim.x + threadIdx.x; i < NYg * NXg; i += blockDim.x * 64)
    m = fmaxf(m, base[i]);
  __shared__ float sm[256];
  sm[threadIdx.x] = m;
  __syncthreads();
  for (int o = 128; o > 0; o >>= 1) {
    if ((int)threadIdx.x < o) sm[threadIdx.x] = fmaxf(sm[threadIdx.x], sm[threadIdx.x + o]);
    __syncthreads();
  }
  if (threadIdx.x == 0) atomicMax(&mx[b], __float_as_uint(sm[0]));
}

// ---------------- 2x antialiased linear downsample + per-batch normalize -------
__global__ void k_resize(const float* __restrict__ dm, const unsigned* __restrict__ mx,
                         float* __restrict__ out) {
  int idx = blockIdx.x * blockDim.x + threadIdx.x;
  if (idx >= Bn * HW) return;
  int ox = idx % Wc, oy = (idx / Wc) % Hc, b = idx / HW;
  const float wt[4] = {0.25f, 0.75f, 0.75f, 0.25f};
  float wsy = 0.f, wsx = 0.f;
#pragma unroll
  for (int t = 0; t < 4; ++t) {
    int ry = 2 * oy - 1 + t;
    if (ry >= 0 && ry < NYg) wsy += wt[t];
    int rx = 2 * ox - 1 + t;
    if (rx >= 0 && rx < NXg) wsx += wt[t];
  }
  const float* base = dm + (size_t)b * NYg * NXg;
  float num = 0.f;
#pragma unroll
  for (int t = 0; t < 4; ++t) {
    int ry = 2 * oy - 1 + t;
    if (ry < 0 || ry >= NYg) continue;
    float rv = 0.f;
#pragma unroll
    for (int u = 0; u < 4; ++u) {
      int rx = 2 * ox - 1 + u;
      if (rx >= 0 && rx < NXg) rv += wt[u] * base[(size_t)ry * NXg + rx];
    }
    num += wt[t] * rv;
  }
  float m = __uint_as_float(mx[b]);
  float s = (m > 0.f) ? (1.0f / m) : 1.0f;
  out[idx] = s * num / (wsy * wsx);
}

// ---------------- conv1: 1 -> 8 channels, 3x3, pad 1 ----------------
__global__ void k_conv1(const float* __restrict__ in, const float* __restrict__ w1,
                        float* __restrict__ out) {
  int idx = blockIdx.x * blockDim.x + threadIdx.x;
  if (idx >= Bn * HW) return;
  int x = idx % Wc, y = (idx / Wc) % Hc, b = idx / HW;
  float p[9];
#pragma unroll
  for (int ky = 0; ky < 3; ++ky) {
    int yy = y + ky - 1;
#pragma unroll
    for (int kx = 0; kx < 3; ++kx) {
      int xx = x + kx - 1;
      p[ky * 3 + kx] = (yy >= 0 && yy < Hc && xx >= 0 && xx < Wc)
                           ? in[(size_t)b * HW + (size_t)yy * Wc + xx] : 0.f;
    }
  }
#pragma unroll
  for (int c = 0; c < 8; ++c) {
    float o = 0.f;
#pragma unroll
    for (int k = 0; k < 9; ++k) o += w1[c * 9 + k] * p[k];
    out[((size_t)(b * 8 + c)) * HW + (size_t)y * Wc + x] = o;
  }
}

// ---------------- per-channel sum/sumsq over (B,H,W) ----------------
__global__ void k_stats(const float* __restrict__ x, float* __restrict__ sum,
                        float* __restrict__ sq, int C, int SPLIT) {
  int s = blockIdx.x % SPLIT;
  int pc = blockIdx.x / SPLIT;  // b*C + c
  int c = pc % C;
  const float* base = x + (size_t)pc * HW;
  float ls = 0.f, lq = 0.f;
  for (int i = s * blockDim.x + threadIdx.x; i < HW; i += blockDim.x * SPLIT) {
    float v = base[i];
    ls += v; lq += v * v;
  }
  __shared__ float ssum[256], ssq[256];
  ssum[threadIdx.x] = ls; ssq[threadIdx.x] = lq;
  __syncthreads();
  for (int o = 128; o > 0; o >>= 1) {
    if ((int)threadIdx.x < o) {
      ssum[threadIdx.x] += ssum[threadIdx.x + o];
      ssq[threadIdx.x]  += ssq[threadIdx.x + o];
    }
    __syncthreads();
  }
  if (threadIdx.x == 0) { atomicAdd(&sum[c], ssum[0]); atomicAdd(&sq[c], ssq[0]); }
}

// ---------------- training-mode BN (biased var) + ReLU ----------------
__global__ void k_bnrelu(const float* __restrict__ x, float* __restrict__ y,
                         const float* __restrict__ sum, const float* __restrict__ sq,
                         const float* __restrict__ gamma, const float* __restrict__ beta,
                         int C, size_t outBatchStride, size_t outOff) {
  int idx = blockIdx.x * blockDim.x + threadIdx.x;
  if (idx >= Bn * C * HW) return;
  int pix = idx % HW;
  int pc = idx / HW;
  int c = pc % C, b = pc / C;
  float invN = 1.0f / (float)(Bn * HW);
  float mean = sum[c] * invN;
  float var = sq[c] * invN - mean * mean;
  float is = rsqrtf(var + BNEPS);
  float v = (x[idx] - mean) * is * gamma[c] + beta[c];
  y[outOff + (size_t)b * outBatchStride + (size_t)c * HW + pix] = fmaxf(v, 0.f);
}

// ---------------- conv2: 8 -> 16, 3x3 as GEMM via V_WMMA_F32_16X16X4_F32 -------
// D[16 cout x 16 px] = A[16 x 72] * B[72 x 16], K = cin*9 + tap, 18 K-steps of 4.
// LDS-staged: each block owns one (b,y) output row per iteration; the 8ch x 3row
// zero-padded halo lives in LDS so the MAC loop is branch-free and low-pressure.
// Wave32 fragment layout (ISA 7.12.2): lanes 0-15 hold K=k0+{0,1}, lanes 16-31
// hold K=k0+{2,3}; D vreg r: lanes 0-15 M=r, lanes 16-31 M=r+8.
__global__ void __launch_bounds__(256) k_conv2_wmma(const float* __restrict__ h1,
                                                    const float* __restrict__ w2,
                                                    float* __restrict__ out2) {
  __shared__ float sw[16 * 72];       // A matrix (weights)
  __shared__ int   soff[72];          // im2col LDS-offset LUT per K
  __shared__ float st[24 * LDSW];     // staged tile: (c*3+ky) rows, cols = gx+1

  for (int i = threadIdx.x; i < 16 * 72; i += 256) sw[i] = w2[i];
  for (int i = threadIdx.x; i < 72; i += 256) {
    int c = i / 9, tap = i % 9;
    soff[i] = (c * 3 + tap / 3) * LDSW + (tap % 3);
  }
  __syncthreads();

  const int lane = threadIdx.x & 31;
  const int wv = threadIdx.x >> 5;    // wave in block: 0..7
  const int m = lane & 15;
  const int hi = lane >> 4;           // 0: lanes 0-15, 1: lanes 16-31
  const int khalf = hi << 1;

  const int nrows = Bn * Hc;          // 992 output rows
  for (int row = blockIdx.x; row < nrows; row += gridDim.x) {
    int b = row / Hc, y = row % Hc;
    // stage zero-padded halo: rows (c,ky), LDS col = global x + 1
    for (int i = threadIdx.x; i < 24 * LCOLS; i += 256) {
      int r = i / LCOLS, col = i % LCOLS;
      int c = r / 3, ky = r % 3;
      int yy = y + ky - 1;
      int xx = col - 1;
      float v = 0.f;
      if (yy >= 0 && yy < Hc && xx >= 0 && xx < Wc)
        v = h1[((size_t)(b * 8 + c)) * HW + (size_t)yy * Wc + xx];
      st[r * LDSW + col] = v;
    }
    __syncthreads();

    for (int t = wv; t < TILES_X; t += 8) {
      int xb = t * 16 + m;            // lane's pixel column (may exceed Wc-1 in last tile)
      v8f acc = {};
#pragma unroll
      for (int s = 0; s < 18; ++s) {
        int k0 = 4 * s + khalf;
        v2f av, bf;
        av.x = sw[m * 72 + k0];
        av.y = sw[m * 72 + k0 + 1];
        bf[0] = st[soff[k0] + xb];
        bf[1] = st[soff[k0 + 1] + xb];
        acc = __builtin_amdgcn_wmma_f32_16x16x4_f32(false, av, false, bf,
                                                    (short)0, acc, false, false);
      }
#pragma unroll
      for (int rr = 0; rr < 8; ++rr) {
        int mm = rr + (hi << 3);
        if (xb < Wc)
          out2[((size_t)(b * 16 + mm)) * HW + (size_t)y * Wc + xb] = acc[rr];
      }
    }
    __syncthreads();
  }
}

// ---------------- concat copy: spatial (B,384,H,W) -> out channels [0,384) -----
__global__ void k_concat(const float* __restrict__ in, float* __restrict__ out) {
  constexpr size_t HW4 = HW / 4;                    // 13392
  constexpr size_t TOT4 = (size_t)Bn * CIN * HW4;   // 20,570,112
  size_t idx = blockIdx.x * (size_t)blockDim.x + threadIdx.x;
  if (idx >= TOT4) return;
  size_t b = idx / ((size_t)CIN * HW4);
  const f4* ip = (const f4*)in;
  f4* op = (f4*)out;
  f4 v = __builtin_nontemporal_load(&ip[idx]);
  __builtin_nontemporal_store(v, &op[idx + b * (size_t)NDF * HW4]);
}

extern "C" void kernel_launch(void* const* d_in, const int* in_sizes, int n_in,
                              void* d_out, int out_size, void* d_ws, size_t ws_size,
                              hipStream_t stream) {
  const float* spatial = (const float*)d_in[0];
  const float* points  = (const float*)d_in[1];
  const float* w1      = (const float*)d_in[2];
  const float* gamma1  = (const float*)d_in[3];
  const float* beta1   = (const float*)d_in[4];
  const float* w2      = (const float*)d_in[5];
  const float* gamma2  = (const float*)d_in[6];
  const float* beta2   = (const float*)d_in[7];
  float* out = (float*)d_out;
  float* ws  = (float*)d_ws;

  const int T = 256;
  // 1) zero header + histogram; gaussian weights
  k_zero<<<(int)((HIST_O + GRID3 + T - 1) / T), T, 0, stream>>>(ws);
  k_gauss_weights<<<1, 1, 0, stream>>>(ws + GW_O);
  // 2) point histogram
  k_hist<<<(NPTS + T - 1) / T, T, 0, stream>>>(points, ws + HIST_O);
  // 3) separable blur
  int gGrid = (int)((GRID3 + T - 1) / T);
  k_blur_y<<<gGrid, T, 0, stream>>>(ws + HIST_O, ws + TMP_O, ws + GW_O);
  k_blur_x<<<gGrid, T, 0, stream>>>(ws + TMP_O, ws + DM_O, ws + GW_O);
  // 4) per-batch max
  k_max<<<Bn * 64, T, 0, stream>>>(ws + DM_O, (unsigned*)(ws + MAX_O));
  // 5) normalize + 2x antialiased downsample
  k_resize<<<(Bn * HW + T - 1) / T, T, 0, stream>>>(ws + DM_O, (const unsigned*)(ws + MAX_O),
                                                    ws + DMR_O);
  // 6) conv1 -> stats -> BN+ReLU
  k_conv1<<<(Bn * HW + T - 1) / T, T, 0, stream>>>(ws + DMR_O, w1, ws + OUT1_O);
  k_stats<<<Bn * 8 * 8, T, 0, stream>>>(ws + OUT1_O, ws + ST1S_O, ws + ST1Q_O, 8, 8);
  k_bnrelu<<<(Bn * 8 * HW + T - 1) / T, T, 0, stream>>>(ws + OUT1_O, ws + H1_O,
                                                        ws + ST1S_O, ws + ST1Q_O,
                                                        gamma1, beta1, 8,
                                                        (size_t)8 * HW, 0);
  // 7) conv2 via WMMA -> stats -> BN+ReLU directly into d_out channel block [384,400)
  k_conv2_wmma<<<496, 256, 0, stream>>>(ws + H1_O, w2, ws + OUT2_O);
  k_stats<<<Bn * 16 * 8, T, 0, stream>>>(ws + OUT2_O, ws + ST2S_O, ws + ST2Q_O, 16, 8);
  k_bnrelu<<<(Bn * 16 * HW + T - 1) / T, T, 0, stream>>>(ws + OUT2_O, out,
                                                         ws + ST2S_O, ws + ST2Q_O,
                                                         gamma2, beta2, 16,
                                                         (size_t)(CIN + NDF) * HW,
                                                         (size_t)CIN * HW);
  // 8) bandwidth-dominant concat copy (NT hints, B128)
  size_t tot4 = (size_t)Bn * CIN * (HW / 4);
  k_concat<<<(int)((tot4 + T - 1) / T), T, 0, stream>>>(spatial, out);
}